// MPC_65034394796826
// MI455X (gfx1250) — compile-verified
//
#include <hip/hip_runtime.h>

// MPC QP -> exact LQR Riccati recursion (backward sweep + forward rollout).
// One wave32 per batch; fp32 WMMA (v_wmma_f32_16x16x4_f32) for the 12-dim
// Schur-complement products. All matrix tiles live in LDS padded to 16x16 so
// WMMA fragment accesses are unconditional (no EXEC-mask ladders); the zero
// padding is preserved by the WMMA math itself.

#define NS_  8
#define NC_  4
#define T_   32
#define NB_  128
#define NSC_ 12   // NS + NC
#define T1_  33   // T + 1

typedef float v2f __attribute__((ext_vector_type(2)));
typedef float v8f __attribute__((ext_vector_type(8)));

__device__ __forceinline__ v8f wmma4(v2f a, v2f b, v8f acc) {
  // D = A(16x4) * B(4x16) + C, fp32 throughout.
  return __builtin_amdgcn_wmma_f32_16x16x4_f32(
      /*neg_a=*/false, a, /*neg_b=*/false, b,
      /*c_mod=*/(short)0, acc, /*reuse_a=*/false, /*reuse_b=*/false);
}

__global__ __launch_bounds__(32) void mpc_riccati(
    const float* __restrict__ x0g,  // [NB][NS]
    const float* __restrict__ Cg_,  // [T][NB][NSC][NSC]
    const float* __restrict__ cg_,  // [T][NB][NSC]
    const float* __restrict__ Fg_,  // [T][NB][NS][NSC]
    const float* __restrict__ fg_,  // [T][NB][NS]
    float* __restrict__ out)        // x:[T1][NB][NS] then u:[T1][NB][NC]
{
  const int  b    = blockIdx.x;
  const int  lane = threadIdx.x;
  const int  hl   = lane & 15;      // half-wave lane
  const bool hiw  = lane >= 16;     // upper half-wave
  const int  rbase = hiw ? 8 : 0;   // D/C-layout row base for this half-wave
  const int  kbase = hiw ? 2 : 0;   // A/B-layout k base for this half-wave

  // 16x16 zero-padded tiles (padding provably stays zero; see notes).
  __shared__ float Pm[16][16];            // value Hessian P_t    (8x8 valid)
  __shared__ float Wm[16][16];            // W = P * F            (8x12 valid)
  __shared__ float Qm[16][16];            // Q = C + F^T P F      (12x12 valid)
  __shared__ float Fs[16][16];            // staged F_t           (8x12 valid)
  __shared__ float Cs[16][16];            // staged C_t           (12x12 valid)
  __shared__ float Km[NC_][16];           // gain K_t             (4x8 valid)
  __shared__ float kvv[NC_];              // affine gain k_t
  __shared__ float pv[NS_];               // value gradient p_t
  __shared__ float qv[NSC_];              // q = c + F^T (P f + p)
  __shared__ float Pfp[NS_];              // P f + p
  __shared__ float Kall[T1_][NC_][NS_];   // stored gains for rollout
  __shared__ float kall[T1_][NC_];
  __shared__ float xs[NS_];
  __shared__ float us[NC_];
  __shared__ float xn[NS_];

  // ---- zero padded tiles once
  for (int idx = lane; idx < 256; idx += 32) {
    int r = idx >> 4, cx = idx & 15;
    Pm[r][cx] = 0.f; Wm[r][cx] = 0.f; Qm[r][cx] = 0.f;
    Fs[r][cx] = 0.f; Cs[r][cx] = 0.f;
    if (r < NC_) Km[r][cx] = 0.f;
  }
  __syncthreads();

  // ---- solve Quu, form gains, update (P, p). Qm/qv valid on entry.
  auto solve_update = [&](int t) {
    // 4x4 SPD inverse, redundantly on every lane (tiny; avoids serialization)
    float A[4][4], Inv[4][4];
    #pragma unroll
    for (int i = 0; i < 4; ++i)
      #pragma unroll
      for (int j = 0; j < 4; ++j) {
        A[i][j]   = Qm[NS_ + i][NS_ + j];
        Inv[i][j] = (i == j) ? 1.f : 0.f;
      }
    #pragma unroll
    for (int cc = 0; cc < 4; ++cc) {
      float piv = 1.f / A[cc][cc];
      #pragma unroll
      for (int j = 0; j < 4; ++j) { A[cc][j] *= piv; Inv[cc][j] *= piv; }
      #pragma unroll
      for (int r = 0; r < 4; ++r) {
        if (r == cc) continue;
        float fac = A[r][cc];
        #pragma unroll
        for (int j = 0; j < 4; ++j) { A[r][j] -= fac * A[cc][j]; Inv[r][j] -= fac * Inv[cc][j]; }
      }
    }
    // K = -Quu^{-1} Qux : 4x8 = one entry per lane (all 32 lanes used)
    {
      int jj = lane >> 3;   // 0..3
      int ii = lane & 7;    // 0..7
      float s = 0.f;
      #pragma unroll
      for (int m = 0; m < 4; ++m) s += Inv[jj][m] * Qm[NS_ + m][ii];
      Km[jj][ii]      = -s;
      Kall[t][jj][ii] = -s;
    }
    if (lane < NC_) {   // k = -Quu^{-1} q_u
      float s = 0.f;
      #pragma unroll
      for (int m = 0; m < 4; ++m) s += Inv[lane][m] * qv[NS_ + m];
      kvv[lane]     = -s;
      kall[t][lane] = -s;
    }
    __syncthreads();
    // p_new = q_x + Q_xu k
    if (lane < NS_) {
      float s = qv[lane];
      #pragma unroll
      for (int j = 0; j < NC_; ++j) s += Qm[lane][NS_ + j] * kvv[j];
      pv[lane] = s;
    }
    // P_new = Q_xx + Q_xu * K : single 16x16x4 f32 WMMA.
    // Loads are in-bounds; masking done with selects (v_cndmask), not EXEC.
    v2f a, bb;
    {
      float a0 = Qm[hl][NS_ + kbase];
      float a1 = Qm[hl][NS_ + kbase + 1];
      a.x  = (hl < NS_) ? a0 : 0.f;     // A = Q_xu (8x4)
      a.y  = (hl < NS_) ? a1 : 0.f;
      bb.x = Km[kbase][hl];             // B = K (4x8, zero-padded cols)
      bb.y = Km[kbase + 1][hl];
    }
    v8f cf;
    #pragma unroll
    for (int r = 0; r < 8; ++r) {
      int row = r + rbase;
      float v = Qm[row][hl];
      cf[r] = (row < NS_ && hl < NS_) ? v : 0.f;   // C = Q_xx
    }
    v8f d = wmma4(a, bb, cf);
    __syncthreads();
    #pragma unroll
    for (int r = 0; r < 8; ++r) Pm[r + rbase][hl] = d[r];  // padding stays 0
    __syncthreads();
  };

  // ================= terminal step t = T: Q = C[T-1], q = c[T-1] ==========
  {
    const float* Cg = Cg_ + (size_t)((T_ - 1) * NB_ + b) * NSC_ * NSC_;
    for (int idx = lane; idx < NSC_ * NSC_; idx += 32)
      Qm[idx / NSC_][idx % NSC_] = Cg[idx];
    const float* cg = cg_ + (size_t)((T_ - 1) * NB_ + b) * NSC_;
    if (lane < NSC_) qv[lane] = cg[lane];
    __syncthreads();
    solve_update(T_);
  }

  // ================= backward sweep t = T-1 .. 0 ==========================
  for (int t = T_ - 1; t >= 0; --t) {
    const float* Fg = Fg_ + (size_t)(t * NB_ + b) * NS_ * NSC_;
    const float* fg = fg_ + (size_t)(t * NB_ + b) * NS_;
    const float* Cg = Cg_ + (size_t)(t * NB_ + b) * NSC_ * NSC_;
    const float* cg = cg_ + (size_t)(t * NB_ + b) * NSC_;

    if (t > 0) {  // prefetch next step's tiles (global_prefetch_b8)
      __builtin_prefetch(Fg_ + (size_t)((t - 1) * NB_ + b) * NS_ * NSC_, 0, 0);
      __builtin_prefetch(Cg_ + (size_t)((t - 1) * NB_ + b) * NSC_ * NSC_, 0, 0);
    }

    // ---- stage F (8x12) and C (12x12) into zero-padded LDS (coalesced)
    for (int idx = lane; idx < NS_ * NSC_; idx += 32)
      Fs[idx / NSC_][idx % NSC_] = Fg[idx];
    for (int idx = lane; idx < NSC_ * NSC_; idx += 32)
      Cs[idx / NSC_][idx % NSC_] = Cg[idx];
    __syncthreads();

    // F fragments: serve as WMMA1's B (F as KxN) and WMMA2's A (F^T as MxK);
    // both index F[k][m] identically in this layout.
    v2f fb0, fb1;
    fb0.x = Fs[kbase][hl];     fb0.y = Fs[kbase + 1][hl];
    fb1.x = Fs[4 + kbase][hl]; fb1.y = Fs[4 + kbase + 1][hl];

    // ---- W = P * F : M=8,K=8,N=12 -> two chained k=4 WMMAs (unconditional)
    v8f acc = {0.f, 0.f, 0.f, 0.f, 0.f, 0.f, 0.f, 0.f};
    {
      v2f a0, a1;
      a0.x = Pm[hl][kbase];     a0.y = Pm[hl][kbase + 1];
      a1.x = Pm[hl][4 + kbase]; a1.y = Pm[hl][4 + kbase + 1];
      acc = wmma4(a0, fb0, acc);
      acc = wmma4(a1, fb1, acc);
    }
    #pragma unroll
    for (int r = 0; r < 8; ++r) Wm[r + rbase][hl] = acc[r];  // padding stays 0

    // ---- Pf + p (lanes 0..7)
    if (lane < NS_) {
      float s = pv[lane];
      #pragma unroll
      for (int m = 0; m < NS_; ++m) s += Pm[lane][m] * fg[m];
      Pfp[lane] = s;
    }
    __syncthreads();

    // ---- Q = C + F^T * W : accumulator seeded with staged C (unconditional)
    v8f acc2;
    #pragma unroll
    for (int r = 0; r < 8; ++r) acc2[r] = Cs[r + rbase][hl];
    {
      v2f b0, b1;
      b0.x = Wm[kbase][hl];     b0.y = Wm[kbase + 1][hl];
      b1.x = Wm[4 + kbase][hl]; b1.y = Wm[4 + kbase + 1][hl];
      acc2 = wmma4(fb0, b0, acc2);
      acc2 = wmma4(fb1, b1, acc2);
    }
    #pragma unroll
    for (int r = 0; r < 8; ++r) Qm[r + rbase][hl] = acc2[r]; // padding stays 0

    // ---- q = c + F^T (P f + p) (lanes 0..11)
    if (lane < NSC_) {
      float s = cg[lane];
      #pragma unroll
      for (int i = 0; i < NS_; ++i) s += Fs[i][lane] * Pfp[i];
      qv[lane] = s;
    }
    __syncthreads();

    solve_update(t);
  }

  // ================= forward rollout ======================================
  float* out_x = out;                           // [T1][NB][NS]
  float* out_u = out + (size_t)T1_ * NB_ * NS_; // [T1][NB][NC]
  if (lane < NS_) xs[lane] = x0g[(size_t)b * NS_ + lane];
  __syncthreads();
  for (int t = 0; t <= T_; ++t) {
    if (lane < NC_) {
      float s = kall[t][lane];
      #pragma unroll
      for (int i = 0; i < NS_; ++i) s += Kall[t][lane][i] * xs[i];
      us[lane] = s;
    }
    __syncthreads();
    if (lane < NS_) out_x[(size_t)t * NB_ * NS_ + b * NS_ + lane] = xs[lane];
    if (lane < NC_) out_u[(size_t)t * NB_ * NC_ + b * NC_ + lane] = us[lane];
    if (t < T_) {
      const float* Fg = Fg_ + (size_t)(t * NB_ + b) * NS_ * NSC_;
      const float* fg = fg_ + (size_t)(t * NB_ + b) * NS_;
      if (lane < NS_) {
        float s = fg[lane];
        #pragma unroll
        for (int j = 0; j < NS_; ++j) s += Fg[lane * NSC_ + j] * xs[j];
        #pragma unroll
        for (int j = 0; j < NC_; ++j) s += Fg[lane * NSC_ + NS_ + j] * us[j];
        xn[lane] = s;
      }
      __syncthreads();
      if (lane < NS_) xs[lane] = xn[lane];
      __syncthreads();
    }
  }
}

extern "C" void kernel_launch(void* const* d_in, const int* in_sizes, int n_in,
                              void* d_out, int out_size, void* d_ws, size_t ws_size,
                              hipStream_t stream) {
  (void)in_sizes; (void)n_in; (void)d_ws; (void)ws_size; (void)out_size;
  const float* x0 = (const float*)d_in[0];
  const float* C  = (const float*)d_in[1];
  const float* c  = (const float*)d_in[2];
  const float* F  = (const float*)d_in[3];
  const float* f  = (const float*)d_in[4];
  float* out = (float*)d_out;
  mpc_riccati<<<NB_, 32, 0, stream>>>(x0, C, c, F, f, out);
}